// PDA_detector_train_69372311765189
// MI455X (gfx1250) — compile-verified
//
#include <hip/hip_runtime.h>

// PDA detector: N=M=32, Q=4, Kd=16384, 8 iterations.
// One wave32 per k; WMMA f32 16x16x4 for the per-k 32x32 matmuls;
// Gauss-Jordan inverse of R in padded LDS (lane-per-column).

#define Nn   32
#define Mm   32
#define Qq   4
#define KD   16384
#define NIT  8
#define WPB  4        // waves per block
#define AUGW 65       // padded row stride of augmented [32 x 64] matrix
#define N0EF 0.05f
#define ESEF 0.5f

typedef __attribute__((ext_vector_type(2))) float v2f;
typedef __attribute__((ext_vector_type(8))) float v8f;

__device__ __forceinline__ void wave_sync() {
  // Per-wave LDS ops are in-order on CDNA5; we only need to stop the
  // compiler/scheduler from reordering LDS accesses across this point.
  asm volatile("" ::: "memory");
  __builtin_amdgcn_wave_barrier();
  asm volatile("" ::: "memory");
}

__global__ __launch_bounds__(WPB * 32) void pda_detector_kernel(
    const float* __restrict__ Y,    // [32, 16384] row-major
    const float* __restrict__ H,    // [32, 32]    row-major  H[n][m]
    const float* __restrict__ Xr,   // [32, 4]     row-major  x[m][q]
    const float* __restrict__ Mu,   // [8]
    float* __restrict__ Out)        // [16384*32, 4]
{
  __shared__ float Hs[Nn * Mm];          // H  [n][m]
  __shared__ float Ht[Mm * Nn];          // H^T [m][n]
  __shared__ float aug[WPB][Nn * AUGW];  // per-wave [R | I] -> [W | Rinv]
  __shared__ float srs[WPB][Mm];
  __shared__ float dls[WPB][Mm];
  __shared__ float yts[WPB][Nn];

  const int tid  = threadIdx.x;
  const int lane = tid & 31;
  const int wv   = tid >> 5;
  const int k    = blockIdx.x * WPB + wv;

  // Stage H and H^T in LDS once per block.
  for (int i = tid; i < Nn * Mm; i += WPB * 32) {
    const float h = H[i];
    const int n = i >> 5, m = i & 31;
    Hs[i] = h;
    Ht[m * Nn + n] = h;
  }
  __syncthreads();

  float* A   = aug[wv];
  float* SRv = srs[wv];
  float* DLv = dls[wv];
  float* YTv = yts[wv];

  __builtin_prefetch(Y + (size_t)lane * KD + k, 0, 0);   // global_prefetch_b8
  const float y = Y[(size_t)lane * KD + k];              // lane n's received sample
  const float x0 = Xr[lane * Qq + 0];
  const float x1 = Xr[lane * Qq + 1];
  const float x2 = Xr[lane * Qq + 2];
  const float x3 = Xr[lane * Qq + 3];

  float SR = 0.0f, ER = ESEF;
  const int hlf  = lane >> 4;   // 0 | 1 : K-pair selector for WMMA fragments
  const int lrow = lane & 15;   // row/col within a 16-wide tile

  for (int it = 0; it < NIT; ++it) {
    const float mu_i  = Mu[it];
    const float Delta = ER - SR * SR;
    SRv[lane] = SR;
    DLv[lane] = Delta;
    wave_sync();

    // Y_tilde[n] = y[n] - sum_m H[n][m] * SR[m]   (lane plays n)
    float yt = y;
    #pragma unroll
    for (int m = 0; m < Mm; ++m) yt -= Ht[m * Nn + lane] * SRv[m];
    YTv[lane] = yt;
    wave_sync();

    // ---- R = H diag(Delta) H^T + N0*I  via WMMA f32 16x16x4 ----
    // A_frag[i][c] = H[i][c]*Delta[c], B_frag[c][j] = H[j][c]
    #pragma unroll
    for (int ti = 0; ti < 2; ++ti) {
      #pragma unroll
      for (int tj = 0; tj < 2; ++tj) {
        v8f acc = {};
        #pragma unroll
        for (int kk = 0; kk < 8; ++kk) {
          const int c0 = kk * 4 + hlf * 2;
          const int ra = ti * 16 + lrow;
          const int cb = tj * 16 + lrow;
          v2f a, b;
          a.x = Hs[ra * Mm + c0]     * DLv[c0];
          a.y = Hs[ra * Mm + c0 + 1] * DLv[c0 + 1];
          b.x = Hs[cb * Mm + c0];
          b.y = Hs[cb * Mm + c0 + 1];
          acc = __builtin_amdgcn_wmma_f32_16x16x4_f32(
              false, a, false, b, (short)0, acc, false, false);
        }
        #pragma unroll
        for (int v = 0; v < 8; ++v) {
          const int r = ti * 16 + hlf * 8 + v;
          const int c = tj * 16 + lrow;
          A[r * AUGW + c] = acc[v] + ((r == c) ? N0EF : 0.0f);
        }
      }
    }
    // Augment with identity in columns 32..63.
    #pragma unroll
    for (int i = 0; i < Nn; ++i)
      A[i * AUGW + 32 + lane] = (i == lane) ? 1.0f : 0.0f;
    wave_sync();

    // ---- Gauss-Jordan: [R | I] -> [I | Rinv]  (lane owns cols L, L+32) ----
    for (int p = 0; p < Nn; ++p) {
      const float pivinv = 1.0f / A[p * AUGW + p];   // broadcast LDS read
      const float r0 = A[p * AUGW + lane] * pivinv;
      const float r1 = A[p * AUGW + 32 + lane] * pivinv;
      A[p * AUGW + lane]      = r0;
      A[p * AUGW + 32 + lane] = r1;
      wave_sync();
      for (int i = 0; i < Nn; ++i) {
        if (i == p) continue;
        const float f = A[i * AUGW + p];             // read before col-p update
        A[i * AUGW + lane]      -= f * r0;
        A[i * AUGW + 32 + lane] -= f * r1;
      }
      wave_sync();
    }

    // ---- tmp = H^T * Rinv  via WMMA; store into cols 0..31 of aug ----
    // A_frag[m][i] = H[i][m] = Ht[m][i], B_frag[i][n] = Rinv[i][n]
    #pragma unroll
    for (int ti = 0; ti < 2; ++ti) {
      #pragma unroll
      for (int tj = 0; tj < 2; ++tj) {
        v8f acc = {};
        #pragma unroll
        for (int kk = 0; kk < 8; ++kk) {
          const int c0 = kk * 4 + hlf * 2;
          const int ra = ti * 16 + lrow;   // m
          const int cb = tj * 16 + lrow;   // n
          v2f a, b;
          a.x = Ht[ra * Nn + c0];
          a.y = Ht[ra * Nn + c0 + 1];
          b.x = A[c0 * AUGW + 32 + cb];
          b.y = A[(c0 + 1) * AUGW + 32 + cb];
          acc = __builtin_amdgcn_wmma_f32_16x16x4_f32(
              false, a, false, b, (short)0, acc, false, false);
        }
        #pragma unroll
        for (int v = 0; v < 8; ++v) {
          const int r = ti * 16 + hlf * 8 + v;
          const int c = tj * 16 + lrow;
          A[r * AUGW + c] = acc[v];        // cols < 32: disjoint from Rinv reads
        }
      }
    }
    wave_sync();

    // ---- Gamma[m] = sum_n W[m][n]*H[n][m];  t2[m] = sum_n W[m][n]*yt[n] ----
    float G = 0.0f, T = 0.0f;
    #pragma unroll
    for (int n = 0; n < Nn; ++n) {
      const float w = A[lane * AUGW + n];
      G += w * Ht[lane * Nn + n];
      T += w * YTv[n];
    }

    const float coef = mu_i / (1.0f - G * Delta);
    const float base = T + G * SR;
    float L0 = coef * (x0 * base - 0.5f * x0 * x0 * G);
    float L1 = coef * (x1 * base - 0.5f * x1 * x1 * G);
    float L2 = coef * (x2 * base - 0.5f * x2 * x2 * G);
    float L3 = coef * (x3 * base - 0.5f * x3 * x3 * G);
    const float mx = fmaxf(fmaxf(L0, L1), fmaxf(L2, L3));

    if (it < NIT - 1) {
      // Softmax over Q=4 candidates, then posterior mean/energy.
      const float e0 = __expf(L0 - mx);
      const float e1 = __expf(L1 - mx);
      const float e2 = __expf(L2 - mx);
      const float e3 = __expf(L3 - mx);
      const float inv = 1.0f / (e0 + e1 + e2 + e3);
      SR = (x0 * e0 + x1 * e1 + x2 * e2 + x3 * e3) * inv;
      ER = (x0 * x0 * e0 + x1 * x1 * e1 + x2 * x2 * e2 + x3 * x3 * e3) * inv;
      wave_sync();
    } else {
      float4 o = make_float4(L0 - mx, L1 - mx, L2 - mx, L3 - mx);
      *(float4*)(Out + ((size_t)k * Mm + lane) * Qq) = o;   // global_store_b128
    }
  }
}

extern "C" void kernel_launch(void* const* d_in, const int* in_sizes, int n_in,
                              void* d_out, int out_size, void* d_ws, size_t ws_size,
                              hipStream_t stream) {
  (void)in_sizes; (void)n_in; (void)out_size; (void)d_ws; (void)ws_size;
  const float* Y  = (const float*)d_in[0];
  const float* H  = (const float*)d_in[1];
  const float* Xr = (const float*)d_in[2];
  const float* Mu = (const float*)d_in[3];
  float* Out = (float*)d_out;

  dim3 grid(KD / WPB);
  dim3 block(WPB * 32);
  hipLaunchKernelGGL(pda_detector_kernel, grid, block, 0, stream, Y, H, Xr, Mu, Out);
}